// Bone_Direction_GCN_60249801228392
// MI455X (gfx1250) — compile-verified
//
#include <hip/hip_runtime.h>
#include <hip/hip_bf16.h>

// ---------------- problem constants ----------------
#define B_TOT   16384
#define JNT     17
#define E_NUM   32
#define CIN     128
#define COUTC   128
#define MIDC    64
#define SLOPE   0.01f

// ---------------- tiling ----------------
#define TB      8                 // batch elements per block
#define ROWS    (TB * JNT)        // 136 live rows
#define RT      9                 // row tiles (144 padded rows)
#define PROWS   (RT * 16)         // 144

// LDS leading dims (floats), chosen for bank-conflict-free fragment reads
#define XLD     132
#define HLD     68
#define W1LD    132
#define W2LD    68
#define W4LD    132
#define ZLD     132
#define GLD     36                // padded mixing K (17 + 17 + 2 zero)

// LDS layout (float offsets)
#define OFF_X   0                         // 144*132 = 19008
#define OFF_H   (OFF_X  + PROWS * XLD)    // 144*68  =  9792
#define OFF_W1  (OFF_H  + PROWS * HLD)    // 128*132 = 16896
#define OFF_W2  (OFF_W1 + CIN * W1LD)     // 128*68  =  8704
#define OFF_W4  (OFF_W2 + CIN * W2LD)     // 64*132  =  8448
#define OFF_Z   (OFF_W4 + MIDC * W4LD)    // 36*132  =  4752
#define OFF_G   (OFF_Z  + GLD * ZLD)      // 32*36   =  1152
#define OFF_RS  (OFF_G  + 32 * GLD)       // 20
#define OFF_B2  (OFF_RS + 20)             // 64
#define SMEM_FLOATS (OFF_B2 + 64)
#define SMEM_BYTES  (SMEM_FLOATS * 4)     // 275,344 B < 320 KB

typedef __attribute__((ext_vector_type(2))) float v2f;
typedef __attribute__((ext_vector_type(8))) float v8f;

// ---------------------------------------------------------------------------
// Tensor Data Mover staging (gfx1250): one DMA per 2D tile, global -> LDS,
// with descriptor-side row padding 128 dwords -> +4 dwords (= 132-float LD).
// Guarded so host pass / toolchains without the builtin fall back cleanly.
// ---------------------------------------------------------------------------
#if defined(__gfx1250__) && __has_builtin(__builtin_amdgcn_tensor_load_to_lds)
#define USE_TDM 1
typedef __attribute__((ext_vector_type(4))) unsigned int u32x4_t;
typedef __attribute__((ext_vector_type(4))) int          i32x4_t;
typedef __attribute__((ext_vector_type(8))) int          i32x8_t;

// Load `rows` x 128 f32 tile (row-major, stride 128) from gptr into LDS at
// lds_byte_off with LDS row stride 132 floats (pad_interval=128dw, pad=4dw).
__device__ __forceinline__ void tdm_load_rows128(const void* gptr,
                                                 unsigned lds_byte_off,
                                                 int rows) {
  unsigned long long ga = (unsigned long long)gptr;
  u32x4_t g0;
  g0[0] = 1u;                                            // count=1 (valid D#)
  g0[1] = lds_byte_off;                                  // lds_addr
  g0[2] = (unsigned)(ga & 0xFFFFFFFFu);                  // global_addr lo
  g0[3] = (unsigned)((ga >> 32) & 0x01FFFFFFu) | (2u << 30); // addr hi | type=2
  i32x8_t g1;
  g1[0] = (int)((2u << 16)      // data_size = 4B
              | (1u << 20)      // pad_enable
              | (6u << 22)      // pad_interval: 128 DWORDs
              | (3u << 25));    // pad_amount : 4 DWORDs
  g1[1] = (int)(128u << 16);                 // tensor_dim0 = 128 (lo16 @63:48)
  g1[2] = (int)((unsigned)rows << 16);       // tensor_dim1 lo16 @95:80
  g1[3] = (int)(128u << 16);                 // tile_dim0 = 128 @127:112
  g1[4] = rows;                              // tile_dim1 @143:128
  g1[5] = 128;                               // tensor_dim0_stride lo32
  g1[6] = 0;
  g1[7] = 0;                                 // tensor_dim1_stride (unused, 2D)
  i32x4_t z4 = {0, 0, 0, 0};
#if __clang_major__ >= 23
  i32x8_t z8 = {0, 0, 0, 0, 0, 0, 0, 0};
  __builtin_amdgcn_tensor_load_to_lds(g0, g1, z4, z4, z8, 0);
#else
  __builtin_amdgcn_tensor_load_to_lds(g0, g1, z4, z4, 0);
#endif
}
#endif

// D = A(16x4,f32) * B(4x16,f32) + C(16x16,f32)
__device__ __forceinline__ v8f wmma4(v2f a, v2f b, v8f c) {
  return __builtin_amdgcn_wmma_f32_16x16x4_f32(
      /*neg_a=*/false, a, /*neg_b=*/false, b,
      /*c_mod=*/(short)0, c, /*reuse_a=*/false, /*reuse_b=*/false);
}

// A fragment: element (vgpr v, half h) = A[M = lane%16][k = v + 2h]
// contiguous + 8B aligned (ld even, k0 multiple of 4) -> single b64 LDS load
__device__ __forceinline__ v2f ldA(const float* S, int row0, int k0, int ld,
                                   int h, int r) {
  return *(const v2f*)(S + (row0 + r) * ld + k0 + 2 * h);
}
// B fragment: element (vgpr v, half h) = B[k = v + 2h][N = lane%16]
__device__ __forceinline__ v2f ldB(const float* S, int k0, int c0, int ld,
                                   int h, int r) {
  v2f b;
  b.x = S[(k0 + 2 * h    ) * ld + c0 + r];
  b.y = S[(k0 + 2 * h + 1) * ld + c0 + r];
  return b;
}

// ---------------------------------------------------------------------------
// Prep: build G[32][36] = [ M_gcn (17x17) | 0.5*adj^T (17x17) | 0 ] and
// colsum(adj)[17] into workspace. Trivial work, one lane.
// ---------------------------------------------------------------------------
__global__ void prep_kernel(const int* __restrict__ ei,
                            const float* __restrict__ ew,
                            const float* __restrict__ adj,
                            float* __restrict__ Gws) {
  if (threadIdx.x != 0 || blockIdx.x != 0) return;
  float deg[JNT], dis[JNT], Mm[JNT][JNT];
  for (int j = 0; j < JNT; ++j) {
    deg[j] = 1.0f;                       // self-loop weight
    for (int i = 0; i < JNT; ++i) Mm[j][i] = 0.0f;
  }
  for (int e = 0; e < E_NUM; ++e) deg[ei[E_NUM + e]] += ew[e];
  for (int j = 0; j < JNT; ++j) dis[j] = deg[j] > 0.f ? rsqrtf(deg[j]) : 0.f;
  for (int e = 0; e < E_NUM; ++e) {
    int rr = ei[e], cc = ei[E_NUM + e];
    Mm[cc][rr] += dis[rr] * ew[e] * dis[cc];
  }
  for (int j = 0; j < JNT; ++j) Mm[j][j] += dis[j] * dis[j];

  for (int jo = 0; jo < 32; ++jo)
    for (int k = 0; k < GLD; ++k) {
      float val = 0.f;
      if (jo < JNT) {
        if (k < JNT)            val = Mm[jo][k];
        else if (k < 2 * JNT)   val = 0.5f * adj[(k - JNT) * JNT + jo];
      }
      Gws[jo * GLD + k] = val;
    }
  for (int jo = 0; jo < JNT; ++jo) {
    float s = 0.f;
    for (int v = 0; v < JNT; ++v) s += adj[v * JNT + jo];
    Gws[32 * GLD + jo] = s;
  }
}

// ---------------------------------------------------------------------------
// Fused main kernel:
// out = x + M@(x W1) + b1 + 0.5*(adj^T @ (lrelu(x W2^T+b2) W4^T + b4))
// ---------------------------------------------------------------------------
__global__ __launch_bounds__(256)
void fused_kernel(const float* __restrict__ xg_all,
                  const float* __restrict__ W1, const float* __restrict__ b1,
                  const float* __restrict__ W2, const float* __restrict__ b2,
                  const float* __restrict__ W4, const float* __restrict__ b4,
                  const float* __restrict__ Gws,
                  float* __restrict__ out) {
  extern __shared__ float sm[];
  float* Xs  = sm + OFF_X;
  float* Hs  = sm + OFF_H;
  float* W1s = sm + OFF_W1;
  float* W2t = sm + OFF_W2;   // W2^T : [k][m]
  float* W4t = sm + OFF_W4;   // W4^T : [m][c]
  float* Zb  = sm + OFF_Z;    // [36][8*16] joint-major staging for mixing GEMM
  float* Gs  = sm + OFF_G;
  float* rss = sm + OFF_RS;
  float* b2s = sm + OFF_B2;

  const int tid = threadIdx.x;
  const int wv = tid >> 5, lane = tid & 31, h = lane >> 4, r = lane & 15;
  const int b0 = blockIdx.x * TB;
  const float* xg = xg_all + (size_t)b0 * (JNT * CIN);

  // ---- stage 0: staging into LDS ----
#ifdef USE_TDM
  // X tile and W1 via Tensor Data Mover (one DMA each, descriptor-side
  // 128->132 row padding); issued once by wave 0, EXEC-independent.
  if (wv == 0) {
    unsigned base = (unsigned)__builtin_amdgcn_groupstaticsize();
    tdm_load_rows128(xg, base + OFF_X * 4, ROWS);
    tdm_load_rows128(W1, base + OFF_W1 * 4, CIN);
  }
#else
  for (int i = tid; i < ROWS * (CIN / 4); i += 256) {
    int rr = i >> 5, c4 = (i & 31) << 2;
    float4 v = *(const float4*)(xg + rr * CIN + c4);
    *(float4*)(Xs + rr * XLD + c4) = v;
  }
  for (int i = tid; i < (CIN * COUTC) / 4; i += 256) {
    int rr = i >> 5, c4 = (i & 31) << 2;
    float4 v = *(const float4*)(W1 + rr * COUTC + c4);
    *(float4*)(W1s + rr * W1LD + c4) = v;
  }
#endif
  for (int i = tid; i < (PROWS - ROWS) * XLD; i += 256) Xs[ROWS * XLD + i] = 0.f;
  for (int i = tid; i < MIDC * CIN; i += 256) {       // W2[m][k] -> W2t[k][m]
    int m = i >> 7, k = i & 127;
    W2t[k * W2LD + m] = W2[i];
  }
  for (int i = tid; i < COUTC * MIDC; i += 256) {     // W4[c][m] -> W4t[m][c]
    int c = i >> 6, m = i & 63;
    W4t[m * W4LD + c] = W4[i];
  }
  for (int i = tid; i < 32 * GLD; i += 256) Gs[i] = Gws[i];
  if (tid < JNT) rss[tid] = Gws[32 * GLD + tid];
  if (tid < MIDC) b2s[tid] = b2[tid];
  for (int i = tid; i < 2 * ZLD; i += 256) Zb[34 * ZLD + i] = 0.f; // pad K rows
  if (tid == 0) __builtin_prefetch(xg + TB * JNT * CIN, 0, 1);
#ifdef USE_TDM
  if (wv == 0) __builtin_amdgcn_s_wait_tensorcnt(0);  // DMA done before barrier
#endif
  __syncthreads();

  // ---- GEMM1: H = leaky_relu(X @ W2^T + b2), 136x64 over K=128 ----
  for (int t = wv; t < RT * (MIDC / 16); t += 8) {     // 36 C-tiles
    int row0 = (t >> 2) * 16, m0 = (t & 3) * 16;
    v8f acc = {};
    for (int k0 = 0; k0 < CIN; k0 += 4)
      acc = wmma4(ldA(Xs, row0, k0, XLD, h, r),
                  ldB(W2t, k0, m0, W2LD, h, r), acc);
    float bb = b2s[m0 + r];
#pragma unroll
    for (int v = 0; v < 8; ++v) {
      float val = acc[v] + bb;
      val = val >= 0.f ? val : SLOPE * val;
      Hs[(row0 + v + 8 * h) * HLD + m0 + r] = val;
    }
  }
  __syncthreads();

  // ---- per 16-channel slab: Y1 = X@W1, Y4 = H@W4^T -> Z; then G @ Z ----
  for (int nc = 0; nc < COUTC / 16; ++nc) {
    const int c0 = nc * 16;
    for (int t = wv; t < RT; t += 8) {                 // Y1 row tiles
      int row0 = t * 16;
      v8f acc = {};
      for (int k0 = 0; k0 < CIN; k0 += 4)
        acc = wmma4(ldA(Xs, row0, k0, XLD, h, r),
                    ldB(W1s, k0, c0, W1LD, h, r), acc);
#pragma unroll
      for (int v = 0; v < 8; ++v) {
        int grow = row0 + v + 8 * h;
        int elem = grow / JNT, j = grow - elem * JNT;
        if (elem < TB) Zb[j * ZLD + elem * 16 + r] = acc[v];
      }
    }
    for (int t = wv; t < RT; t += 8) {                 // Y4 row tiles
      int row0 = t * 16;
      v8f acc = {};
      for (int k0 = 0; k0 < MIDC; k0 += 4)
        acc = wmma4(ldA(Hs, row0, k0, HLD, h, r),
                    ldB(W4t, k0, c0, W4LD, h, r), acc);
#pragma unroll
      for (int v = 0; v < 8; ++v) {
        int grow = row0 + v + 8 * h;
        int elem = grow / JNT, j = grow - elem * JNT;
        if (elem < TB) Zb[(JNT + j) * ZLD + elem * 16 + r] = acc[v];
      }
    }
    __syncthreads();

    // mixing GEMM: OUT(17x(8*16)) = G(17x36) @ Z(36x(8*16)) + epilogue
    for (int t = wv; t < 16; t += 8) {
      int mt = t >> 3, nt = t & 7;                     // nt == batch elem
      int row0 = mt * 16, zc0 = nt * 16;
      v8f acc = {};
      for (int k0 = 0; k0 < GLD; k0 += 4)
        acc = wmma4(ldA(Gs, row0, k0, GLD, h, r),
                    ldB(Zb, k0, zc0, ZLD, h, r), acc);
      int cc = c0 + r;
      float b1c = b1[cc], b4c = b4[cc];
#pragma unroll
      for (int v = 0; v < 8; ++v) {
        int jo = row0 + v + 8 * h;
        if (jo < JNT) {
          float res = Xs[(nt * JNT + jo) * XLD + cc] + acc[v] + b1c +
                      0.5f * rss[jo] * b4c;
          out[(((size_t)(b0 + nt)) * JNT + jo) * COUTC + cc] = res;
        }
      }
    }
    __syncthreads();
  }
}

// ---------------------------------------------------------------------------
extern "C" void kernel_launch(void* const* d_in, const int* in_sizes, int n_in,
                              void* d_out, int out_size, void* d_ws,
                              size_t ws_size, hipStream_t stream) {
  const float* vec = (const float*)d_in[0];
  const int*   ei  = (const int*)d_in[1];
  const float* ew  = (const float*)d_in[2];
  const float* W1  = (const float*)d_in[3];
  const float* b1  = (const float*)d_in[4];
  const float* W2  = (const float*)d_in[5];
  const float* b2  = (const float*)d_in[6];
  const float* W4  = (const float*)d_in[7];
  const float* b4  = (const float*)d_in[8];
  const float* adj = (const float*)d_in[9];
  float* out = (float*)d_out;
  float* Gws = (float*)d_ws;   // 32*36 + 17 floats

  prep_kernel<<<1, 32, 0, stream>>>(ei, ew, adj, Gws);
  fused_kernel<<<B_TOT / TB, 256, SMEM_BYTES, stream>>>(
      vec, W1, b1, W2, b2, W4, b4, Gws, out);
}